// GIN_14697378087406
// MI455X (gfx1250) — compile-verified
//
#include <hip/hip_runtime.h>
#include <hip/hip_bf16.h>

// GIN forward for MI455X (gfx1250, wave32).
// - FP32 WMMA (v_wmma_f32_16x16x4_f32) for node-feature GEMMs, 2 row tiles
//   per wave (grid = N/32), branch-free specializations via template.
// - BN(scale/shift)+ReLU fused into the A-operand of the following GEMM,
//   with the 256-float scale/shift vector staged in LDS.
// - (1+eps)*h fused into the scatter-destination init; edge/node indices
//   forced scalar (readfirstlane -> s_load); mean-pool refactored as
//   pool(h)@Wl + bl. h/agg buffers (51.2MB) are L2-resident (192MB L2),
//   so the scatter phase runs at L2 bandwidth.

typedef __attribute__((ext_vector_type(2))) float v2f;
typedef __attribute__((ext_vector_type(8))) float v8f;

#define HDIM 128
#define GDIM 512
#define TDIM 10
#define BN_EPS 1e-5f

// ---------------------------------------------------------------- utilities
__global__ void zero_f32(float* __restrict__ p, int n) {
    int i = blockIdx.x * blockDim.x + threadIdx.x;
    if (i < n) p[i] = 0.0f;
}

// ------------------------------------------------------------ WMMA GEMM
// Y[m, n] = f(A)[m, :] @ W[:, n] + bias[n]
// TRANS=false: f = identity.  TRANS=true: f(a_k) = relu(a_k*sc[k] + sc[128+k])
// Block = 256 threads = 8 waves (one per 16-col tile); each wave computes
// two 16x16 tiles (rows m0..m0+31). gridDim.x = rows/32.
template <bool TRANS>
__global__ __launch_bounds__(256) void gemm128_wmma(
    const float* __restrict__ A, const float* __restrict__ W,
    const float* __restrict__ bias, const float* __restrict__ scsh,
    float* __restrict__ Y)
{
    const int wave = threadIdx.x >> 5;     // 0..7 -> column tile
    const int lane = threadIdx.x & 31;
    const int half = lane >> 4;            // 0 or 1
    const int l16  = lane & 15;
    const int m0 = blockIdx.x * 32;
    const int n0 = wave * 16;

    __shared__ float sc[2 * HDIM];
    if (TRANS) {
        sc[threadIdx.x] = scsh[threadIdx.x];   // blockDim == 256 == 2*HDIM
        __syncthreads();
    }

    const float* arow0 = A + (size_t)(m0 + l16) * HDIM;
    const float* arow1 = arow0 + (size_t)16 * HDIM;

    v8f acc0 = {}, acc1 = {};
    #pragma unroll 4
    for (int k = 0; k < HDIM; k += 4) {
        const int ka = k + 2 * half;       // this lane's K pair: ka, ka+1
        v2f bv;
        bv[0] = W[(size_t)ka * HDIM + n0 + l16];
        bv[1] = W[(size_t)(ka + 1) * HDIM + n0 + l16];
        float2 a0 = *(const float2*)(arow0 + ka);
        float2 a1 = *(const float2*)(arow1 + ka);
        if (TRANS) {
            const float s0 = sc[ka], s1 = sc[ka + 1];
            const float t0 = sc[HDIM + ka], t1 = sc[HDIM + ka + 1];
            a0.x = fmaxf(fmaf(a0.x, s0, t0), 0.0f);
            a0.y = fmaxf(fmaf(a0.y, s1, t1), 0.0f);
            a1.x = fmaxf(fmaf(a1.x, s0, t0), 0.0f);
            a1.y = fmaxf(fmaf(a1.y, s1, t1), 0.0f);
        }
        v2f av0; av0[0] = a0.x; av0[1] = a0.y;
        v2f av1; av1[0] = a1.x; av1[1] = a1.y;
        acc0 = __builtin_amdgcn_wmma_f32_16x16x4_f32(
            false, av0, false, bv, (short)0, acc0, false, false);
        acc1 = __builtin_amdgcn_wmma_f32_16x16x4_f32(
            false, av1, false, bv, (short)0, acc1, false, false);
    }

    const float bn = bias[n0 + l16];
    #pragma unroll
    for (int i = 0; i < 8; ++i) {
        // C/D layout: VGPR i -> row (i + 8*half), col = l16
        Y[(size_t)(m0 + i + 8 * half) * HDIM + n0 + l16]      = acc0[i] + bn;
        Y[(size_t)(m0 + 16 + i + 8 * half) * HDIM + n0 + l16] = acc1[i] + bn;
    }
}

// ------------------------------------------------------- BN column stats
// stats[0:128] = column sums, stats[128:256] = column sums of squares.
__global__ __launch_bounds__(128) void colstats(
    const float* __restrict__ Y, float* __restrict__ stats,
    int rows, int rowsPerBlock)
{
    const int c  = threadIdx.x;           // 0..127
    int r0 = blockIdx.x * rowsPerBlock;
    int r1 = r0 + rowsPerBlock; if (r1 > rows) r1 = rows;
    float s = 0.0f, ss = 0.0f;
    for (int r = r0; r < r1; ++r) {
        float v = Y[(size_t)r * HDIM + c];
        s += v; ss += v * v;
    }
    atomicAdd(&stats[c], s);
    atomicAdd(&stats[HDIM + c], ss);
}

__global__ __launch_bounds__(128) void bn_finalize(
    const float* __restrict__ stats, const float* __restrict__ g,
    const float* __restrict__ be, float* __restrict__ scsh, float invN)
{
    const int c = threadIdx.x;
    float mu  = stats[c] * invN;
    float var = stats[HDIM + c] * invN - mu * mu;
    float sc  = g[c] * rsqrtf(var + BN_EPS);
    scsh[c] = sc;
    scsh[HDIM + c] = be[c] - mu * sc;
}

// h = relu(bn(y)) in place (float4 vectorized; HDIM multiple of 4)
__global__ __launch_bounds__(256) void apply_bn_relu(
    float* __restrict__ Y, const float* __restrict__ scsh, int n4)
{
    int i = blockIdx.x * blockDim.x + threadIdx.x;
    if (i >= n4) return;
    int c = (i * 4) & (HDIM - 1);
    float4 v = ((float4*)Y)[i];
    v.x = fmaxf(fmaf(v.x, scsh[c],     scsh[HDIM + c]),     0.0f);
    v.y = fmaxf(fmaf(v.y, scsh[c + 1], scsh[HDIM + c + 1]), 0.0f);
    v.z = fmaxf(fmaf(v.z, scsh[c + 2], scsh[HDIM + c + 2]), 0.0f);
    v.w = fmaxf(fmaf(v.w, scsh[c + 3], scsh[HDIM + c + 3]), 0.0f);
    ((float4*)Y)[i] = v;
}

// dst = (1 + eps[l-1]) * h  (scatter-destination init; eps read on device)
__global__ __launch_bounds__(256) void init_gin(
    float* __restrict__ dst, const float* __restrict__ h,
    const float* __restrict__ eps, int l, int n4)
{
    int i = blockIdx.x * blockDim.x + threadIdx.x;
    if (i >= n4) return;
    float s = 1.0f + eps[l - 1];
    float4 v = ((const float4*)h)[i];
    v.x *= s; v.y *= s; v.z *= s; v.w *= s;
    ((float4*)dst)[i] = v;
}

// acc[dst[e], :] += h[src[e], :]  — one wave per edge, float4 per lane.
// Edge id is wave-uniform: force it scalar so index loads become s_load.
__global__ __launch_bounds__(256) void scatter_edges(
    const int* __restrict__ ei, const float* __restrict__ h,
    float* __restrict__ acc, int E_)
{
    int tid  = blockIdx.x * blockDim.x + threadIdx.x;
    int e    = __builtin_amdgcn_readfirstlane(tid >> 5);
    if (e >= E_) return;                       // uniform branch
    int lane = tid & 31;
    int s = ei[e];
    int d = ei[E_ + e];
    float4 v = *(const float4*)(h + (size_t)s * HDIM + lane * 4);
    float* p = acc + (size_t)d * HDIM + lane * 4;
    atomicAdd(p + 0, v.x);
    atomicAdd(p + 1, v.y);
    atomicAdd(p + 2, v.z);
    atomicAdd(p + 3, v.w);
}

// per-graph sums + counts — one wave per node, float4 per lane
__global__ __launch_bounds__(256) void pool_accum(
    const float* __restrict__ h, const int* __restrict__ batch,
    float* __restrict__ pool, float* __restrict__ pcnt, int rows)
{
    int tid  = blockIdx.x * blockDim.x + threadIdx.x;
    int n    = __builtin_amdgcn_readfirstlane(tid >> 5);
    if (n >= rows) return;                     // uniform branch
    int lane = tid & 31;
    int g = batch[n];
    float4 v = *(const float4*)(h + (size_t)n * HDIM + lane * 4);
    float* p = pool + (size_t)g * HDIM + lane * 4;
    atomicAdd(p + 0, v.x);
    atomicAdd(p + 1, v.y);
    atomicAdd(p + 2, v.z);
    atomicAdd(p + 3, v.w);
    if (lane == 0) atomicAdd(&pcnt[g], 1.0f);
}

// out accumulation: mean_pool(h) @ Wl + bl
// first==1 (layer 0): out = cnt>0 ? mean@Wl+bl : 0   (== mean_pool(h@Wl+bl))
// first==0:           out += mean@Wl + bl
__global__ __launch_bounds__(256) void pool_project(
    const float* __restrict__ pool, const float* __restrict__ pcnt,
    const float* __restrict__ Wl, const float* __restrict__ bl,
    float* __restrict__ oacc, int first)
{
    int tid = blockIdx.x * blockDim.x + threadIdx.x;
    if (tid >= GDIM * TDIM) return;
    int g = tid / TDIM, t = tid % TDIM;
    float cnt = pcnt[g];
    float inv = 1.0f / fmaxf(cnt, 1.0f);
    float s = 0.0f;
    for (int k = 0; k < HDIM; ++k)
        s += pool[(size_t)g * HDIM + k] * Wl[(size_t)k * TDIM + t];
    s = s * inv + bl[t];
    if (first) oacc[tid] = (cnt > 0.0f) ? s : 0.0f;
    else       oacc[tid] += s;
}

__global__ __launch_bounds__(256) void sigmoid_k(
    const float* __restrict__ oacc, float* __restrict__ out, int n)
{
    int i = blockIdx.x * blockDim.x + threadIdx.x;
    if (i < n) out[i] = 1.0f / (1.0f + __expf(-oacc[i]));
}

// ---------------------------------------------------------------- driver
extern "C" void kernel_launch(void* const* d_in, const int* in_sizes, int n_in,
                              void* d_out, int out_size, void* d_ws, size_t ws_size,
                              hipStream_t stream) {
    const float* x     = (const float*)d_in[0];
    const int*   ei    = (const int*)d_in[1];
    const int*   batch = (const int*)d_in[2];
    const float* Wa    = (const float*)d_in[3];
    const float* ba    = (const float*)d_in[4];
    const float* ga    = (const float*)d_in[5];
    const float* bea   = (const float*)d_in[6];
    const float* Wb    = (const float*)d_in[7];
    const float* bb    = (const float*)d_in[8];
    const float* gb    = (const float*)d_in[9];
    const float* beb   = (const float*)d_in[10];
    const float* Wl    = (const float*)d_in[11];
    const float* bl    = (const float*)d_in[12];
    const float* eps   = (const float*)d_in[13];
    float*       out   = (float*)d_out;

    const int Nn = in_sizes[0] / HDIM;       // 100000
    const int Ee = in_sizes[1] / 2;          // 1600000
    const size_t NH = (size_t)Nn * HDIM;

    float* ws    = (float*)d_ws;
    float* bufA  = ws;                        // N*H
    float* bufB  = ws + NH;                   // N*H
    float* stats = ws + 2 * NH;               // 256
    float* scsh  = stats + 256;               // 256
    float* pool  = scsh + 256;                // G*H
    float* pcnt  = pool + (size_t)GDIM * HDIM;// G
    float* oacc  = pcnt + GDIM;               // G*T

    const int   mtiles  = Nn / 32;            // 3125 (2 row tiles per wave)
    const float invN    = 1.0f / (float)Nn;
    const int   n4      = (int)(NH / 4);
    const int   eb4     = (n4 + 255) / 256;
    const int   csBlk   = 256;                // rows per colstats block
    const int   csGrid  = (Nn + csBlk - 1) / csBlk;
    const int   edgeGrid = (Ee * 32 + 255) / 256;
    const int   nodeGrid = (Nn * 32 + 255) / 256;
    const int   projGrid = (GDIM * TDIM + 255) / 256;

    auto run_bn = [&](const float* Y, const float* g, const float* be) {
        zero_f32<<<1, 256, 0, stream>>>(stats, 256);
        colstats<<<csGrid, 128, 0, stream>>>(Y, stats, Nn, csBlk);
        bn_finalize<<<1, 128, 0, stream>>>(stats, g, be, scsh, invN);
    };
    auto run_pool = [&](const float* h, const float* Wl_l, const float* bl_l,
                        int first) {
        zero_f32<<<(GDIM * HDIM + GDIM + 255) / 256, 256, 0, stream>>>(
            pool, GDIM * HDIM + GDIM);        // pool + pcnt contiguous
        pool_accum<<<nodeGrid, 256, 0, stream>>>(h, batch, pool, pcnt, Nn);
        pool_project<<<projGrid, 256, 0, stream>>>(pool, pcnt, Wl_l, bl_l,
                                                   oacc, first);
    };

    // ---- layer 0: h = MLP(x); out = mean_pool(h @ Wl0 + bl0)
    gemm128_wmma<false><<<mtiles, 256, 0, stream>>>(x, Wa, ba, nullptr, bufB);
    run_bn(bufB, ga, bea);
    gemm128_wmma<true><<<mtiles, 256, 0, stream>>>(bufB, Wb, bb, scsh, bufA);
    run_bn(bufA, gb, beb);
    apply_bn_relu<<<eb4, 256, 0, stream>>>(bufA, scsh, n4);
    run_pool(bufA, Wl, bl, 1);

    float* h = bufA;   // current node features
    float* t = bufB;   // scratch / ping-pong

    // ---- layers 1..3: GINConv + MLP + pooled projection
    for (int l = 1; l < 4; ++l) {
        const float* Wa_l = Wa + (size_t)l * HDIM * HDIM;
        const float* Wb_l = Wb + (size_t)l * HDIM * HDIM;
        init_gin<<<eb4, 256, 0, stream>>>(t, h, eps, l, n4);       // t=(1+eps)h
        scatter_edges<<<edgeGrid, 256, 0, stream>>>(ei, h, t, Ee); // t+=agg
        gemm128_wmma<false><<<mtiles, 256, 0, stream>>>(t, Wa_l, ba + l * HDIM,
                                                        nullptr, h); // y1 -> h
        run_bn(h, ga + l * HDIM, bea + l * HDIM);
        gemm128_wmma<true><<<mtiles, 256, 0, stream>>>(h, Wb_l, bb + l * HDIM,
                                                       scsh, t);     // y2 -> t
        run_bn(t, gb + l * HDIM, beb + l * HDIM);
        apply_bn_relu<<<eb4, 256, 0, stream>>>(t, scsh, n4);       // h' in t
        run_pool(t, Wl + (size_t)l * HDIM * TDIM, bl + (size_t)l * TDIM, 0);
        float* tmp = h; h = t; t = tmp;
    }

    sigmoid_k<<<projGrid, 256, 0, stream>>>(oacc, out, GDIM * TDIM);
}